// TernarySpikeActivation_17712445129311
// MI455X (gfx1250) — compile-verified
//
#include <hip/hip_runtime.h>
#include <stdint.h>

// ---------------------------------------------------------------------------
// TernarySpikeActivation forward (16-step LIF, subtract reset, ternary spike,
// mean over time). Purely elementwise -> no WMMA; CDNA5 path used is the
// async global->LDS engine (ASYNCcnt) with a double-buffered pipeline and
// non-temporal output stores.
// ---------------------------------------------------------------------------

#define SPK_STEPS    16
#define TILE_THREADS 256
#define MAX_BLOCKS   4096

typedef float v4f __attribute__((ext_vector_type(4)));

// Issue one per-lane 16-byte async copy global -> LDS (GVS addressing:
// 64-bit SGPR base + 32-bit per-lane byte offset). Tracked by ASYNCcnt.
__device__ __forceinline__ void async_f4_to_lds(const float* base,
                                                unsigned byte_off,
                                                unsigned lds_byte_addr) {
  asm volatile("global_load_async_to_lds_b128 %0, %1, %2"
               :
               : "v"(lds_byte_addr), "v"(byte_off), "s"(base)
               : "memory");
}

// 16-step LIF recurrence for one element; returns mean ternary spike.
// Per step: 1 v_cndmask (select a vs a-1 using previous mem>1 compare),
// 1 v_fma_f32, 2 v_cmp, 2 integer accumulates.
__device__ __forceinline__ float lif_mean_spk(float a) {
  const float am1 = a - 1.0f;   // hoisted: a - reset*RESET_TH when reset==1
  float mem = 0.0f;
  int pos = 0, neg = 0;
  bool reset = false;
#pragma unroll
  for (int s = 0; s < SPK_STEPS; ++s) {
    float inp = reset ? am1 : a;            // a - reset*1.0
    mem = __builtin_fmaf(0.9f, mem, inp);   // beta*mem + (a - reset)
    bool gt = mem > 1.0f;                   // spike+  (also next reset)
    bool lt = mem < -1.0f;                  // spike-
    pos += gt;
    neg += lt;
    reset = gt;
  }
  return (float)(pos - neg) * (1.0f / (float)SPK_STEPS);
}

__global__ void __launch_bounds__(TILE_THREADS)
spike_stream_kernel(const float* __restrict__ act, float* __restrict__ out,
                    int n4, int ntiles) {
  __shared__ __align__(16) v4f lbuf[2][TILE_THREADS];

  const int tid = threadIdx.x;
  // Byte offsets of this lane's two LDS slots (low 32 bits of the flat
  // shared-aperture address == workgroup-relative LDS byte address).
  const unsigned lds0 = (unsigned)(uintptr_t)&lbuf[0][tid];
  const unsigned lds1 = (unsigned)(uintptr_t)&lbuf[1][tid];

  int tile = blockIdx.x;
  const int stride = gridDim.x;
  if (tile >= ntiles) return;               // uniform per block

  // Prologue: stage tile 0 of this block into buffer 0.
  {
    unsigned idx4 = (unsigned)tile * TILE_THREADS + (unsigned)tid;
    if (idx4 < (unsigned)n4)
      async_f4_to_lds(act, idx4 * 16u, lds0);
  }

  int buf = 0;                              // uniform
  for (; tile < ntiles; tile += stride) {
    const int next = tile + stride;         // uniform
    if (next < ntiles) {
      // Prefetch next tile into the other buffer, then wait for current.
      unsigned nidx4 = (unsigned)next * TILE_THREADS + (unsigned)tid;
      if (nidx4 < (unsigned)n4)
        async_f4_to_lds(act, nidx4 * 16u, (buf == 0) ? lds1 : lds0);
      asm volatile("s_wait_asynccnt 0x1" ::: "memory");
    } else {
      asm volatile("s_wait_asynccnt 0x0" ::: "memory");
    }
    __syncthreads();   // all waves' async data for `tile` is in LDS

    unsigned idx4 = (unsigned)tile * TILE_THREADS + (unsigned)tid;
    if (idx4 < (unsigned)n4) {
      v4f v = lbuf[buf][tid];               // ds_load_b128
      v4f r;
      r.x = lif_mean_spk(v.x);
      r.y = lif_mean_spk(v.y);
      r.z = lif_mean_spk(v.z);
      r.w = lif_mean_spk(v.w);
      // Write-once output: non-temporal 128-bit store.
      __builtin_nontemporal_store(r, (v4f*)out + idx4);
    }
    __syncthreads();   // safe to overwrite lbuf[buf] next iteration
    buf ^= 1;
  }
}

// Scalar tail for n % 4 != 0 (not hit for 4096*8192, but keep it general).
__global__ void spike_tail_kernel(const float* __restrict__ act,
                                  float* __restrict__ out, int start, int n) {
  int i = start + blockIdx.x * blockDim.x + threadIdx.x;
  if (i < n) out[i] = lif_mean_spk(act[i]);
}

extern "C" void kernel_launch(void* const* d_in, const int* in_sizes, int n_in,
                              void* d_out, int out_size, void* d_ws, size_t ws_size,
                              hipStream_t stream) {
  (void)n_in; (void)d_ws; (void)ws_size; (void)out_size;
  const float* act = (const float*)d_in[0];
  float* out = (float*)d_out;

  const int n  = in_sizes[0];          // 4096*8192 = 33,554,432
  const int n4 = n >> 2;               // float4 count
  const int ntiles = (n4 + TILE_THREADS - 1) / TILE_THREADS;

  if (ntiles > 0) {
    int blocks = ntiles < MAX_BLOCKS ? ntiles : MAX_BLOCKS;
    spike_stream_kernel<<<blocks, TILE_THREADS, 0, stream>>>(act, out, n4, ntiles);
  }
  const int tail_start = n4 << 2;
  const int tail = n - tail_start;
  if (tail > 0) {
    spike_tail_kernel<<<(tail + 63) / 64, 64, 0, stream>>>(act, out, tail_start, n);
  }
}